// PairwiseTree_20194936226068
// MI455X (gfx1250) — compile-verified
//
#include <hip/hip_runtime.h>

typedef float v2f __attribute__((ext_vector_type(2)));
typedef float v8f __attribute__((ext_vector_type(8)));

#define T_FULL 2048
#define D      16
#define LEVELS 11
#define WAVES  8

// c19 activation. cs division is replaced by multiply with a per-kernel
// precomputed reciprocal (exact for c=1; <=1ulp otherwise).
__device__ __forceinline__ float c19_act(float xv, float inv_cs, float cs,
                                         float rs, float Lv) {
  float scaled = xv * inv_cs;
  float nf = floorf(scaled);
  float t = scaled - nf;
  float h = t * (1.0f - t);
  int ni = (int)nf;                       // (ni & 1) == Python-parity test
  float hs = ((ni & 1) == 0) ? h : -h;    // sgn * h
  float interior = cs * fmaf(rs * h, h, hs);
  float r = (xv <= -Lv) ? (xv + Lv) : interior;
  r = (xv >= Lv) ? (xv - Lv) : r;
  return r;
}

__global__ __launch_bounds__(256, 1)
void tree_mlp_kernel(const float* __restrict__ x,
                     const float* __restrict__ W1,
                     const float* __restrict__ b1,
                     const float* __restrict__ W2,
                     const float* __restrict__ b2,
                     const float* __restrict__ cpar,
                     const float* __restrict__ rpar,
                     float* __restrict__ out) {
  __shared__ float bufA[1024 * D];          // 64 KB ping
  __shared__ float bufB[512 * D];           // 32 KB pong
  __shared__ float scratch[WAVES][16 * 32]; // 2 KB/wave: input stage + h transpose

  const int n    = blockIdx.x;
  const int tid  = threadIdx.x;
  const int wave = tid >> 5;
  const int lane = tid & 31;
  const int ln   = lane & 15;
  const int hi   = lane >> 4;               // 0: lanes 0-15, 1: lanes 16-31
  const int khi  = hi * 2;

  // ---- preload weight fragments in WMMA B-layout into registers ----
  // B tile (4K x 16N): VGPR v, lane l: element [K = 4k+v+2*(l>=16)][N = l%16]
  v2f w1f[8][2];
  v2f w2f[8];
#pragma unroll
  for (int k = 0; k < 8; ++k) {
#pragma unroll
    for (int v = 0; v < 2; ++v) {
      int K = 4 * k + v + khi;
      w1f[k][0][v] = W1[K * 32 + ln];       // W1 cols 0..15
      w1f[k][1][v] = W1[K * 32 + 16 + ln];  // W1 cols 16..31
      w2f[k][v]    = W2[K * 16 + ln];       // W2 (32x16)
    }
  }
  // per-lane channel parameters (channel = 16*tile + lane%16 in C/D layout)
  const float b1v0 = b1[ln],                b1v1 = b1[16 + ln];
  const float cs0  = fmaxf(cpar[ln], 0.1f), cs1  = fmaxf(cpar[16 + ln], 0.1f);
  const float rs0  = fmaxf(rpar[ln], 0.0f), rs1  = fmaxf(rpar[16 + ln], 0.0f);
  const float L0   = 6.0f * cs0,            L1   = 6.0f * cs1;
  const float ic0  = 1.0f / cs0,            ic1  = 1.0f / cs1;  // once per kernel
  const float b2v  = b2[ln];

  float* scr = &scratch[wave][0];
  const float* xn = x + (size_t)n * T_FULL * D;

  int Pin = T_FULL;
  for (int lev = 0; lev < LEVELS; ++lev) {
    const int Pout     = Pin >> 1;
    const int numTiles = (Pout + 15) >> 4;
    const float* inB   = (lev & 1) ? bufA : bufB;   // valid for lev >= 1
    float*       outB  = (lev & 1) ? bufB : bufA;
    const bool   last  = (lev == LEVELS - 1);

    for (int tile = wave; tile < numTiles; tile += WAVES) {
      // ---- fetch cat tile (16 pairs x 32 feats = 512 contiguous floats) ----
      const float* catBase;
      if (lev == 0) {
        const float4* src4 = (const float4*)(xn + (size_t)tile * 512);
        float4*       dst4 = (float4*)scr;
#pragma unroll
        for (int it = 0; it < 4; ++it)
          dst4[lane + 32 * it] = src4[lane + 32 * it];
        catBase = scr;                      // wave-private; LDS in-order per wave
      } else {
        catBase = inB + tile * 512;         // pair rows are contiguous in buf
      }

      // ---- GEMM1: (16x32) @ W1(32x32); bias b1 folded into C init ----
      v8f acc0 = {b1v0, b1v0, b1v0, b1v0, b1v0, b1v0, b1v0, b1v0};
      v8f acc1 = {b1v1, b1v1, b1v1, b1v1, b1v1, b1v1, b1v1, b1v1};
#pragma unroll
      for (int k = 0; k < 8; ++k) {
        // A-layout 16x4 f32: lane l -> row l%16, K = 4k + 2*(l>=16) + {0,1}
        v2f a = *(const v2f*)(catBase + ln * 32 + 4 * k + khi);
        acc0 = __builtin_amdgcn_wmma_f32_16x16x4_f32(false, a, false, w1f[k][0],
                                                     (short)0, acc0, false, false);
        acc1 = __builtin_amdgcn_wmma_f32_16x16x4_f32(false, a, false, w1f[k][1],
                                                     (short)0, acc1, false, false);
      }

      // ---- c19 activation, transpose h through wave scratch ----
#pragma unroll
      for (int j = 0; j < 8; ++j) {
        int row = j + 8 * hi;               // C/D layout: row = j + 8*hi, col = ln
        scr[row * 32 + ln]      = c19_act(acc0[j], ic0, cs0, rs0, L0);
        scr[row * 32 + 16 + ln] = c19_act(acc1[j], ic1, cs1, rs1, L1);
      }

      // ---- GEMM2: h(16x32) @ W2(32x16); bias b2 folded into C init ----
      v8f accy = {b2v, b2v, b2v, b2v, b2v, b2v, b2v, b2v};
#pragma unroll
      for (int k = 0; k < 8; ++k) {
        v2f a = *(const v2f*)(scr + ln * 32 + 4 * k + khi);
        accy = __builtin_amdgcn_wmma_f32_16x16x4_f32(false, a, false, w2f[k],
                                                     (short)0, accy, false, false);
      }

      // ---- store y; D-layout rows are coalesced 16-float lines ----
#pragma unroll
      for (int j = 0; j < 8; ++j) {
        int row = tile * 16 + j + 8 * hi;
        if (row < Pout) {
          if (last) out[(size_t)n * D + ln] = accy[j];  // Pout==1 -> row 0 only
          else      outB[row * D + ln]      = accy[j];
        }
      }
    }
    __syncthreads();
    Pin = Pout;
  }
}

extern "C" void kernel_launch(void* const* d_in, const int* in_sizes, int n_in,
                              void* d_out, int out_size, void* d_ws, size_t ws_size,
                              hipStream_t stream) {
  const float* x   = (const float*)d_in[0];
  const float* W1  = (const float*)d_in[1];
  const float* b1  = (const float*)d_in[2];
  const float* W2  = (const float*)d_in[3];
  const float* b2  = (const float*)d_in[4];
  const float* c   = (const float*)d_in[5];
  const float* rho = (const float*)d_in[6];
  float* out = (float*)d_out;
  (void)d_ws; (void)ws_size; (void)n_in; (void)out_size;

  int N = in_sizes[0] / (T_FULL * D);       // 4096
  tree_mlp_kernel<<<dim3(N), dim3(256), 0, stream>>>(x, W1, b1, W2, b2, c, rho, out);
}